// MambaForecaster_34471407517780
// MI455X (gfx1250) — compile-verified
//
#include <hip/hip_runtime.h>
#include <hip/hip_bf16.h>
#include <math.h>

typedef float v2f __attribute__((ext_vector_type(2)));
typedef float v8f __attribute__((ext_vector_type(8)));

#define BB     4
#define LL     2048
#define FF     20
#define DM     256
#define NN     64
#define KK     4
#define DIN    512
#define HH     8
#define PP     64
#define DPROJ  1160
#define CCH    640          // Din + 2N
#define NSEG   32
#define SEGLEN 64
#define ROWS   (BB*LL)      // 8192

__device__ __forceinline__ float silu_f(float v){ return v / (1.f + expf(-v)); }

// ---------------------------------------------------------------------------
// Kernel 1: encoder GEMM  h0 = x(8192x20) @ enc_W(20x256) + enc_b
// fp32 WMMA 16x16x4, K=20 -> 5 k-steps. Block: 8 waves = 128 rows x 16 cols.
// ---------------------------------------------------------------------------
__global__ void enc_gemm_kernel(const float* __restrict__ x,
                                const float* __restrict__ W,
                                const float* __restrict__ bias,
                                float* __restrict__ h0) {
  int tid  = threadIdx.x;
  int lane = tid & 31, wid = tid >> 5;
  int m0 = blockIdx.x * 128 + wid * 16;
  int n0 = blockIdx.y * 16;
  int mr = lane & 15;               // M (for A) / N (for B,C)
  int kh = (lane >> 4) << 1;        // lanes 16-31 handle K+2,K+3
  v8f c = {};
  #pragma unroll
  for (int k0 = 0; k0 < FF; k0 += 4) {
    v2f a, bm;
    a.x  = x[(m0 + mr) * FF + k0 + kh];
    a.y  = x[(m0 + mr) * FF + k0 + kh + 1];
    bm.x = W[(k0 + kh) * DM + n0 + mr];
    bm.y = W[(k0 + kh + 1) * DM + n0 + mr];
    c = __builtin_amdgcn_wmma_f32_16x16x4_f32(false, a, false, bm,
                                              (short)0, c, false, false);
  }
  int nn = n0 + mr;
  float bv = bias[nn];
  #pragma unroll
  for (int r = 0; r < 8; ++r) {
    int m = m0 + r + ((lane >> 4) << 3);
    h0[m * DM + nn] = c[r] + bv;
  }
}

// ---------------------------------------------------------------------------
// Kernel 2: LayerNorm over Dm, in place. One row per block, 256 threads.
// ---------------------------------------------------------------------------
__global__ void ln_kernel(float* __restrict__ h0,
                          const float* __restrict__ g,
                          const float* __restrict__ b) {
  __shared__ float red[DM];
  int row = blockIdx.x, tid = threadIdx.x;
  float v = h0[row * DM + tid];
  red[tid] = v; __syncthreads();
  for (int o = DM / 2; o > 0; o >>= 1) { if (tid < o) red[tid] += red[tid + o]; __syncthreads(); }
  float mu = red[0] / DM;
  __syncthreads();
  float d = v - mu;
  red[tid] = d * d; __syncthreads();
  for (int o = DM / 2; o > 0; o >>= 1) { if (tid < o) red[tid] += red[tid + o]; __syncthreads(); }
  float var = red[0] / DM;
  h0[row * DM + tid] = d * rsqrtf(var + 1e-5f) * g[tid] + b[tid];
}

// ---------------------------------------------------------------------------
// Kernel 3: in_proj GEMM restricted to xBC columns [512,1152) -> 640 cols.
// (8192x256)@(256x640), fp32 WMMA 16x16x4.
// Each wave computes a 32x16 tile (two M-tiles sharing one strided B
// fragment -> halves strided-B traffic per FLOP). 8 waves = 2(Mpair)x4(N)
// -> 64x64 block tile. Grid (128, 10). K loop 64, unroll 8 = 16 wmma/body.
// ---------------------------------------------------------------------------
__global__ void inproj_gemm_kernel(const float* __restrict__ h0,
                                   const float* __restrict__ W,
                                   float* __restrict__ xbc) {
  int tid  = threadIdx.x;
  int lane = tid & 31, wid = tid >> 5;
  int wm = wid & 1, wn = wid >> 1;         // 2 x 4 waves
  int m0 = blockIdx.x * 64 + wm * 32;      // wave covers rows m0..m0+31
  int n0 = blockIdx.y * 64 + wn * 16;      // column within [0,640)
  int mr = lane & 15;
  int kh = (lane >> 4) << 1;
  const float* arow0 = h0 + (size_t)(m0 + mr) * DM;
  const float* arow1 = arow0 + (size_t)16 * DM;
  const float* bcol  = W + DIN + n0 + mr;  // skip z columns; stride DPROJ
  v8f c0 = {}, c1 = {};
  #pragma unroll 8
  for (int k0 = 0; k0 < DM; k0 += 4) {
    v2f a0 = *(const v2f*)(arow0 + k0 + kh);   // 8B aligned (even k index)
    v2f a1 = *(const v2f*)(arow1 + k0 + kh);
    v2f bm;
    bm.x = bcol[(size_t)(k0 + kh) * DPROJ];
    bm.y = bcol[(size_t)(k0 + kh + 1) * DPROJ];
    c0 = __builtin_amdgcn_wmma_f32_16x16x4_f32(false, a0, false, bm,
                                               (short)0, c0, false, false);
    c1 = __builtin_amdgcn_wmma_f32_16x16x4_f32(false, a1, false, bm,
                                               (short)0, c1, false, false);
  }
  #pragma unroll
  for (int r = 0; r < 8; ++r) {
    int m = m0 + r + ((lane >> 4) << 3);
    xbc[(size_t)m * CCH + n0 + mr]        = c0[r];
    xbc[(size_t)(m + 16) * CCH + n0 + mr] = c1[r];
  }
}

// ---------------------------------------------------------------------------
// Kernel 4: dt columns (last 8 of in_proj) + softplus + dA = exp(dt*A).
// ---------------------------------------------------------------------------
__global__ void dt_kernel(const float* __restrict__ h0,
                          const float* __restrict__ W,
                          const float* __restrict__ dt_bias,
                          const float* __restrict__ A_log,
                          float* __restrict__ dts, float* __restrict__ dab) {
  int idx = blockIdx.x * blockDim.x + threadIdx.x;  // ROWS*HH
  int row = idx >> 3, hh = idx & 7;
  const float* a = h0 + (size_t)row * DM;
  const float* w = W + (DPROJ - HH) + hh;
  float acc = 0.f;
  for (int k = 0; k < DM; ++k) acc += a[k] * w[(size_t)k * DPROJ];
  float v  = acc + dt_bias[hh];
  float dt = (v > 20.f) ? v : log1pf(expf(v));
  float A  = -expf(A_log[hh]);
  dts[idx] = dt;
  dab[idx] = expf(dt * A);
}

// ---------------------------------------------------------------------------
// Kernel 5: z columns, only at t = L-1 (4 rows x 512).
// ---------------------------------------------------------------------------
__global__ void zlast_kernel(const float* __restrict__ h0,
                             const float* __restrict__ W,
                             float* __restrict__ zl) {
  int b = blockIdx.x, d = threadIdx.x;   // 512 threads
  const float* a = h0 + (size_t)(b * LL + LL - 1) * DM;
  float acc = 0.f;
  for (int k = 0; k < DM; ++k) acc += a[k] * W[(size_t)k * DPROJ + d];
  zl[b * DIN + d] = acc;
}

// ---------------------------------------------------------------------------
// Kernel 6: depthwise causal conv1d (K=4, left pad 3) + bias + SiLU.
// ---------------------------------------------------------------------------
__global__ void conv_kernel(const float* __restrict__ raw,
                            const float* __restrict__ cw,
                            const float* __restrict__ cb,
                            float* __restrict__ act) {
  int row = blockIdx.x, c = threadIdx.x;
  int b = row / LL, l = row % LL;
  float acc = cb[c];
  #pragma unroll
  for (int k = 0; k < KK; ++k) {
    int t = l - (KK - 1) + k;
    if (t >= 0) acc += cw[c * KK + k] * raw[(size_t)(b * LL + t) * CCH + c];
  }
  act[(size_t)row * CCH + c] = silu_f(acc);
}

// ---------------------------------------------------------------------------
// Kernel 7: segmented scan with double-buffered LDS staging.
// dA is scalar per (b,h,t): hc = dA*hc + (dt*xs_p)*B_n decomposes over 32
// segments of 64 steps. Grid (NSEG, HH, BB); 256 threads own 16 elems each.
// Per step: b128 loads of next step's B/xs into the alternate buffer, one
// barrier (publishes t+1, protects t from overwrite at t+2).
// ---------------------------------------------------------------------------
__global__ void scan_seg_kernel(const float* __restrict__ act,
                                const float* __restrict__ dts,
                                const float* __restrict__ dab,
                                float* __restrict__ X, float* __restrict__ Pp) {
  int s = blockIdx.x, h = blockIdx.y, b = blockIdx.z;
  int tid = threadIdx.x;
  int p = tid >> 2, n0 = (tid & 3) * 16;
  __shared__ __align__(16) float sB[2][NN];
  __shared__ __align__(16) float sxs[2][PP];
  __shared__ float sdA[2], sdt[2];
  float hc[16];
  #pragma unroll
  for (int i = 0; i < 16; ++i) hc[i] = 0.f;
  float pr = 1.f;
  int base = b * LL + s * SEGLEN;
  // preload t = 0 into buffer 0
  {
    const float* rowp = act + (size_t)base * CCH;
    if (tid < 16)       *(float4*)&sB[0][tid * 4]         = *(const float4*)(rowp + DIN + tid * 4);
    else if (tid < 32)  *(float4*)&sxs[0][(tid - 16) * 4] = *(const float4*)(rowp + h * PP + (tid - 16) * 4);
    else if (tid == 32) { sdA[0] = dab[base * HH + h]; sdt[0] = dts[base * HH + h]; }
  }
  __syncthreads();
  for (int tt = 0; tt < SEGLEN; ++tt) {
    int cur = tt & 1, nxt = cur ^ 1;
    if (tt + 1 < SEGLEN) {
      int row = base + tt + 1;
      const float* rowp = act + (size_t)row * CCH;
      if (tid < 16)       *(float4*)&sB[nxt][tid * 4]         = *(const float4*)(rowp + DIN + tid * 4);
      else if (tid < 32)  *(float4*)&sxs[nxt][(tid - 16) * 4] = *(const float4*)(rowp + h * PP + (tid - 16) * 4);
      else if (tid == 32) { sdA[nxt] = dab[row * HH + h]; sdt[nxt] = dts[row * HH + h]; }
    }
    float a  = sdA[cur];
    float cf = sdt[cur] * sxs[cur][p];
    pr *= a;
    #pragma unroll
    for (int i = 0; i < 16; ++i) hc[i] = a * hc[i] + cf * sB[cur][n0 + i];
    __syncthreads();
  }
  int bh = b * HH + h;
  float* xo = X + ((size_t)(bh * NSEG + s)) * (PP * NN) + p * NN + n0;
  #pragma unroll
  for (int i = 0; i < 16; ++i) xo[i] = hc[i];
  if (tid == 0) Pp[bh * NSEG + s] = pr;
}

// ---------------------------------------------------------------------------
// Kernel 8: combine segments: hc = P_s*hc + X_s for s = 0..31.
// ---------------------------------------------------------------------------
__global__ void combine_kernel(const float* __restrict__ X,
                               const float* __restrict__ Pp,
                               float* __restrict__ hcF) {
  int h = blockIdx.x, b = blockIdx.y;
  int bh = b * HH + h;
  int tid = threadIdx.x;          // 256
  int off = tid * 16;
  float hc[16];
  #pragma unroll
  for (int i = 0; i < 16; ++i) hc[i] = 0.f;
  for (int s = 0; s < NSEG; ++s) {
    float ps = Pp[bh * NSEG + s];
    const float* xs = X + ((size_t)(bh * NSEG + s)) * (PP * NN) + off;
    #pragma unroll
    for (int i = 0; i < 16; ++i) hc[i] = ps * hc[i] + xs[i];
  }
  float* o = hcF + (size_t)bh * (PP * NN) + off;
  #pragma unroll
  for (int i = 0; i < 16; ++i) o[i] = hc[i];
}

// ---------------------------------------------------------------------------
// Kernel 9: epilogue at t=L-1 only: y = hc·C + D*xs ; gate with silu(z);
// RMS norm; out_proj (512x256); dec (256x3). One block per batch, 512 thr.
// ---------------------------------------------------------------------------
__global__ void epilogue_kernel(const float* __restrict__ act,
                                const float* __restrict__ hcF,
                                const float* __restrict__ zl,
                                const float* __restrict__ Dp,
                                const float* __restrict__ norm_w,
                                const float* __restrict__ Wout,
                                const float* __restrict__ Wdec,
                                const float* __restrict__ bdec,
                                float* __restrict__ out) {
  int b = blockIdx.x, tid = threadIdx.x;   // 512
  int h = tid >> 6, p = tid & 63;
  __shared__ float sC[NN];
  __shared__ float red[DIN];
  __shared__ float sY[DIN];
  __shared__ float sO[DM];
  int lastrow = b * LL + LL - 1;
  if (tid < NN) sC[tid] = act[(size_t)lastrow * CCH + DIN + NN + tid];
  __syncthreads();
  float xs = act[(size_t)lastrow * CCH + h * PP + p];
  const float* hrow = hcF + (size_t)(b * HH + h) * (PP * NN) + p * NN;
  float y = Dp[h] * xs;
  for (int n = 0; n < NN; ++n) y += hrow[n] * sC[n];
  float z = zl[b * DIN + tid];
  y *= silu_f(z);
  red[tid] = y * y; __syncthreads();
  for (int o = DIN / 2; o > 0; o >>= 1) { if (tid < o) red[tid] += red[tid + o]; __syncthreads(); }
  float rms = rsqrtf(red[0] / DIN + 1e-5f);
  float yn = y * rms * norm_w[tid];
  sY[tid] = yn;
  __syncthreads();
  if (tid < DM) {
    float acc = 0.f;
    for (int d = 0; d < DIN; ++d) acc += sY[d] * Wout[(size_t)d * DM + tid];
    sO[tid] = acc;
  }
  __syncthreads();
  if (tid < 3) {
    float acc = bdec[tid];
    for (int j = 0; j < DM; ++j) acc += sO[j] * Wdec[j * 3 + tid];
    out[b * 3 + tid] = acc;
  }
}

// ---------------------------------------------------------------------------
extern "C" void kernel_launch(void* const* d_in, const int* in_sizes, int n_in,
                              void* d_out, int out_size, void* d_ws, size_t ws_size,
                              hipStream_t stream) {
  (void)in_sizes; (void)n_in; (void)out_size; (void)ws_size;
  const float* x       = (const float*)d_in[0];
  const float* enc_W   = (const float*)d_in[1];
  const float* enc_b   = (const float*)d_in[2];
  const float* ln_g    = (const float*)d_in[3];
  const float* ln_b    = (const float*)d_in[4];
  const float* in_W    = (const float*)d_in[5];
  const float* conv_w  = (const float*)d_in[6];
  const float* conv_b  = (const float*)d_in[7];
  const float* dt_bias = (const float*)d_in[8];
  const float* A_log   = (const float*)d_in[9];
  const float* Dp      = (const float*)d_in[10];
  const float* norm_w  = (const float*)d_in[11];
  const float* out_W   = (const float*)d_in[12];
  const float* dec_W   = (const float*)d_in[13];
  const float* dec_b   = (const float*)d_in[14];
  float* out = (float*)d_out;

  float* ws = (float*)d_ws;
  float* h0     = ws; ws += (size_t)ROWS * DM;               // 2.10M
  float* xbcraw = ws; ws += (size_t)ROWS * CCH;              // 5.24M
  float* xbcact = ws; ws += (size_t)ROWS * CCH;              // 5.24M
  float* dts    = ws; ws += (size_t)ROWS * HH;               // 64K
  float* dab    = ws; ws += (size_t)ROWS * HH;               // 64K
  float* Xbuf   = ws; ws += (size_t)BB * HH * NSEG * PP * NN;// 4.19M
  float* Pbuf   = ws; ws += (size_t)BB * HH * NSEG;          // 1K
  float* hcF    = ws; ws += (size_t)BB * HH * PP * NN;       // 131K
  float* zl     = ws; ws += (size_t)BB * DIN;                // 2K

  enc_gemm_kernel<<<dim3(ROWS / 128, DM / 16), 256, 0, stream>>>(x, enc_W, enc_b, h0);
  ln_kernel<<<ROWS, DM, 0, stream>>>(h0, ln_g, ln_b);
  inproj_gemm_kernel<<<dim3(ROWS / 64, CCH / 64), 256, 0, stream>>>(h0, in_W, xbcraw);
  dt_kernel<<<ROWS * HH / 256, 256, 0, stream>>>(h0, in_W, dt_bias, A_log, dts, dab);
  zlast_kernel<<<BB, DIN, 0, stream>>>(h0, in_W, zl);
  conv_kernel<<<ROWS, CCH, 0, stream>>>(xbcraw, conv_w, conv_b, xbcact);
  scan_seg_kernel<<<dim3(NSEG, HH, BB), 256, 0, stream>>>(xbcact, dts, dab, Xbuf, Pbuf);
  combine_kernel<<<dim3(HH, BB), 256, 0, stream>>>(Xbuf, Pbuf, hcF);
  epilogue_kernel<<<BB, DIN, 0, stream>>>(xbcact, hcF, zl, Dp, norm_w, out_W, dec_W, dec_b, out);
}